// AttentionLayer_48696339202352
// MI455X (gfx1250) — compile-verified
//
#include <hip/hip_runtime.h>
#include <hip/hip_bf16.h>
#include <math.h>

// Problem constants (from reference): B=16, S=2048, D=768, out = attn[:,0,:] -> [16,768] f32.
#define BB 16
#define SS 2048
#define DD 768
#define NORM 0.03608439182435161f   // 1/sqrt(768)
#define SCH 16                      // split-S chunks for weighted row-sum

typedef __attribute__((ext_vector_type(2))) float v2f;
typedef __attribute__((ext_vector_type(8))) float v8f;

// ---------------------------------------------------------------------------
// 16x16-tile fp32 GEMM using V_WMMA_F32_16X16X4_F32 (wave32, one wave / tile).
// C[m][n] = scale * sum_k A[m][k] * B[k][n]
//   BT=true : B[k][n] = W[n][k]  (W row-major [Ntot x K]  -> contiguous float2)
//   BT=false: B[k][n] = W[k][n]  (W row-major [K x Ntot])
// A-matrix 16x4 layout: lane m=lane%16, VGPRs hold K = k+koff, k+koff+1,
// koff = (lane<16 ? 0 : 2). B mirrors the K/K+2 half-wave split at N=lane%16.
// C/D: VGPR r -> (M=r, N=lane) lanes 0-15, (M=r+8, N=lane-16) lanes 16-31.
// ---------------------------------------------------------------------------
template <bool BT>
__global__ void gemm16_wmma(const float* __restrict__ A, long strideA,
                            const float* __restrict__ W,
                            float* __restrict__ C, int Ccols, int K, float scale) {
  const int lane = threadIdx.x;          // blockDim.x == 32, EXEC all ones
  const int n0   = blockIdx.x * 16;
  const int m    = lane & 15;
  const int koff = (lane < 16) ? 0 : 2;

  v8f acc = {};
#pragma unroll 8
  for (int k = 0; k < K; k += 4) {
    v2f a = *(const v2f*)(A + (long)m * strideA + k + koff);
    v2f b;
    if (BT) {
      b = *(const v2f*)(W + (long)(n0 + m) * K + k + koff);
    } else {
      b.x = W[(long)(k + koff)     * Ccols + n0 + m];
      b.y = W[(long)(k + koff + 1) * Ccols + n0 + m];
    }
    acc = __builtin_amdgcn_wmma_f32_16x16x4_f32(
        /*neg_a=*/false, a, /*neg_b=*/false, b,
        /*c_mod=*/(short)0, acc, /*reuse_a=*/false, /*reuse_b=*/false);
  }

  const int nOut  = n0 + m;
  const int mBase = (lane < 16) ? 0 : 8;
#pragma unroll
  for (int r = 0; r < 8; ++r)
    C[(long)(mBase + r) * Ccols + nOut] = acc[r] * scale;
}

// ---------------------------------------------------------------------------
// scores[b][s] = qk[b] . x[b,s,:]   (one wave per key row, float4 coalesced)
// ---------------------------------------------------------------------------
__global__ void scores_kernel(const float* __restrict__ x,
                              const float* __restrict__ qk,
                              float* __restrict__ scores) {
  const int wave = threadIdx.x >> 5;
  const int lane = threadIdx.x & 31;
  const int gw   = blockIdx.x * 8 + wave;      // one wave per (b,s)
  const int b    = gw >> 11;                   // S = 2048 waves per batch
  const int s    = gw & (SS - 1);

  const float* row = x  + ((long)b * SS + s) * DD;
  const float* q   = qk + (long)b * DD;

  float acc = 0.f;
#pragma unroll
  for (int i = 0; i < DD / 128; ++i) {         // 6 iters, float4 per lane
    float4 xv = *(const float4*)(row + i * 128 + lane * 4);
    float4 qv = *(const float4*)(q   + i * 128 + lane * 4);
    acc += xv.x * qv.x + xv.y * qv.y + xv.z * qv.z + xv.w * qv.w;
  }
#pragma unroll
  for (int off = 16; off > 0; off >>= 1) acc += __shfl_xor(acc, off);
  if (lane == 0) scores[(long)b * SS + s] = acc;
}

// ---------------------------------------------------------------------------
// softmax over S per batch (mask provably irrelevant for query row 0)
// ---------------------------------------------------------------------------
__global__ void softmax_kernel(const float* __restrict__ scores,
                               float* __restrict__ attn) {
  __shared__ float red[256];
  const int b = blockIdx.x, t = threadIdx.x;
  const float* sc = scores + (long)b * SS;

  float m = -INFINITY;
  for (int i = t; i < SS; i += 256) m = fmaxf(m, sc[i]);
  red[t] = m; __syncthreads();
  for (int w = 128; w > 0; w >>= 1) { if (t < w) red[t] = fmaxf(red[t], red[t + w]); __syncthreads(); }
  m = red[0]; __syncthreads();

  float sum = 0.f;
  for (int i = t; i < SS; i += 256) sum += expf(sc[i] - m);
  red[t] = sum; __syncthreads();
  for (int w = 128; w > 0; w >>= 1) { if (t < w) red[t] += red[t + w]; __syncthreads(); }
  const float inv = 1.0f / red[0];

  for (int i = t; i < SS; i += 256) attn[(long)b * SS + i] = expf(sc[i] - m) * inv;
}

// ---------------------------------------------------------------------------
// partial[b][c][d] = sum_{s in chunk c} attn[b][s] * x[b,s,d]
// ---------------------------------------------------------------------------
__global__ void wsum_partial_kernel(const float* __restrict__ x,
                                    const float* __restrict__ attn,
                                    float* __restrict__ partial) {
  const int b = blockIdx.x / SCH;
  const int c = blockIdx.x % SCH;
  const int t = threadIdx.x;                   // 256 threads, 3 d-values each
  float a0 = 0.f, a1 = 0.f, a2 = 0.f;
  const int s0 = c * (SS / SCH);
  for (int s = s0; s < s0 + SS / SCH; ++s) {
    const float  a   = attn[(long)b * SS + s];
    const float* row = x + ((long)b * SS + s) * DD;
    a0 += a * row[t];
    a1 += a * row[t + 256];
    a2 += a * row[t + 512];
  }
  float* p = partial + ((long)b * SCH + c) * DD;
  p[t] = a0; p[t + 256] = a1; p[t + 512] = a2;
}

__global__ void wsum_reduce_kernel(const float* __restrict__ partial,
                                   float* __restrict__ wsum) {
  const int idx = blockIdx.x * 256 + threadIdx.x;   // 0..B*D-1
  const int b = idx / DD, d = idx % DD;
  float s = 0.f;
#pragma unroll
  for (int c = 0; c < SCH; ++c) s += partial[((long)b * SCH + c) * DD + d];
  wsum[idx] = s;
}

// ---------------------------------------------------------------------------
extern "C" void kernel_launch(void* const* d_in, const int* in_sizes, int n_in,
                              void* d_out, int out_size, void* d_ws, size_t ws_size,
                              hipStream_t stream) {
  const float* b_in = (const float*)d_in[0];
  // d_in[1] = mask: provably a no-op for the returned query row 0 -> ignored.
  const float* Wq = (const float*)d_in[2];
  const float* Wk = (const float*)d_in[3];
  const float* Wv = (const float*)d_in[4];
  float* out = (float*)d_out;

  float* ws      = (float*)d_ws;
  float* q0      = ws;                       // [16, 768]
  float* qk      = q0   + BB * DD;           // [16, 768]
  float* scores  = qk   + BB * DD;           // [16, 2048]
  float* attn    = scores + BB * SS;         // [16, 2048]
  float* partial = attn + BB * SS;           // [16, SCH, 768]
  float* wsum    = partial + (long)BB * SCH * DD;   // [16, 768]

  // 1) q0[b,e] = sum_d x[b,0,d] * Wq[e,d]      (WMMA, B = Wq^T)
  gemm16_wmma<true ><<<DD / 16, 32, 0, stream>>>(b_in, (long)SS * DD, Wq, q0, DD, DD, 1.0f);
  // 2) qk[b,d] = NORM * sum_e q0[b,e] * Wk[e,d] (WMMA, B = Wk)
  gemm16_wmma<false><<<DD / 16, 32, 0, stream>>>(q0, DD, Wk, qk, DD, DD, NORM);
  // 3) scores[b,s] = qk[b] . x[b,s,:]           (bandwidth pass #1 over b_in)
  scores_kernel<<<BB * SS / 8, 256, 0, stream>>>(b_in, qk, scores);
  // 4) softmax over keys
  softmax_kernel<<<BB, 256, 0, stream>>>(scores, attn);
  // 5) wsum[b,d] = sum_s attn[b,s] * x[b,s,d]   (bandwidth pass #2, L2-resident)
  wsum_partial_kernel<<<BB * SCH, 256, 0, stream>>>(b_in, attn, partial);
  wsum_reduce_kernel<<<BB * DD / 256, 256, 0, stream>>>(partial, wsum);
  // 6) out[b,e] = sum_d wsum[b,d] * Wv[e,d]     (WMMA, B = Wv^T)
  gemm16_wmma<true ><<<DD / 16, 32, 0, stream>>>(wsum, DD, Wv, out, DD, DD, 1.0f);
}